// RotateCCDeformConv2D_5385888989767
// MI455X (gfx1250) — compile-verified
//
#include <hip/hip_runtime.h>

// RotateCC Deformable Conv2D for MI455X (gfx1250, wave32).
// Core contraction: M=B*H*W pixels, N=COUT*4=128, K=CIN*9=288, f32,
// executed with V_WMMA_F32_16X16X4_F32 (72 WMMA steps per 16x16 tile)
// to keep full f32 precision vs the f32 reference.

#define BB   2
#define CIN  32
#define COUT 32
#define HH   256
#define WW   256
#define PADC 1
#define KK   9
#define CK   (CIN * KK)   // 288
#define NOUT (COUT * 4)   // 128
#define SROW 294          // padded LDS row stride in dwords (even -> b64 aligned)

typedef float v2f __attribute__((ext_vector_type(2)));
typedef float v8f __attribute__((ext_vector_type(8)));

// ---------------------------------------------------------------------------
// Kernel 1: rotated weights, laid out B-operand friendly:
//   tw_t[n * CK + c*9 + k],  n = o*4 + r,  rot90^r applied to the 3x3 tap grid.
// rot90 (CCW, numpy): rot^1[i][j]=w[j][2-i]; rot^2[i][j]=w[2-i][2-j]; rot^3[i][j]=w[2-j][i]
// ---------------------------------------------------------------------------
__global__ void prep_tw(const float* __restrict__ weight, float* __restrict__ tw_t) {
    int id = blockIdx.x * blockDim.x + threadIdx.x;
    if (id >= NOUT * CK) return;
    int n  = id / CK, ck = id % CK;
    int c  = ck / KK, k  = ck % KK;
    int ky = k / 3,  kx  = k % 3;
    int o  = n >> 2, r   = n & 3;
    int sy, sx;
    switch (r) {
        case 0:  sy = ky;     sx = kx;     break;
        case 1:  sy = kx;     sx = 2 - ky; break;
        case 2:  sy = 2 - ky; sx = 2 - kx; break;
        default: sy = 2 - kx; sx = ky;     break;
    }
    tw_t[id] = weight[(o * CIN + c) * KK + sy * 3 + sx];
}

// ---------------------------------------------------------------------------
// Kernel 2: 3x3 convs for offsets (18ch) + mask (9ch), per-pixel.
// Applies the reference's fixups: off channels 8,9 -> 0 (center tap),
// mask channel 4 -> sigmoid(0)=0.5. Stores [b][k][h][w] planes.
// ---------------------------------------------------------------------------
__global__ void offmask_conv(const float* __restrict__ x,
                             const float* __restrict__ offw, const float* __restrict__ offb,
                             const float* __restrict__ mskw, const float* __restrict__ mskb,
                             float* __restrict__ off_y, float* __restrict__ off_x,
                             float* __restrict__ msk) {
    int p = blockIdx.x * blockDim.x + threadIdx.x;
    if (p >= BB * HH * WW) return;
    int w = p % WW, h = (p / WW) % HH, b = p / (HH * WW);

    float acc[27];
#pragma unroll
    for (int j = 0; j < 18; j++) acc[j] = offb[j];
#pragma unroll
    for (int j = 0; j < 9; j++)  acc[18 + j] = mskb[j];

    for (int c = 0; c < CIN; c++) {
        const float* xp = x + (size_t)(b * CIN + c) * HH * WW;
        float xv[9];
#pragma unroll
        for (int t = 0; t < 9; t++) {
            int yy = h - 1 + t / 3, xx = w - 1 + t % 3;
            xv[t] = (yy >= 0 && yy < HH && xx >= 0 && xx < WW) ? xp[yy * WW + xx] : 0.f;
        }
#pragma unroll
        for (int j = 0; j < 18; j++) {
            const float* wj = offw + (j * CIN + c) * KK;
#pragma unroll
            for (int t = 0; t < 9; t++) acc[j] += xv[t] * wj[t];
        }
#pragma unroll
        for (int j = 0; j < 9; j++) {
            const float* wj = mskw + (j * CIN + c) * KK;
#pragma unroll
            for (int t = 0; t < 9; t++) acc[18 + j] += xv[t] * wj[t];
        }
    }

    size_t base = (size_t)b * KK * HH * WW + (size_t)h * WW + w;
#pragma unroll
    for (int k = 0; k < KK; k++) {
        float oy = (k == 4) ? 0.f : acc[2 * k];
        float ox = (k == 4) ? 0.f : acc[2 * k + 1];
        float mv = (k == 4) ? 0.5f : 1.f / (1.f + __expf(-acc[18 + k]));
        size_t idx = base + (size_t)k * HH * WW;
        off_y[idx] = oy;
        off_x[idx] = ox;
        msk[idx]   = mv;
    }
}

// ---------------------------------------------------------------------------
// Kernel 3: one block (8 wave32) per 16-pixel strip.
//  Stage 1: cooperative bilinear sampling -> sA[m][ck] in LDS (mask applied).
//  Stage 2: wave wv owns N-tile [16*wv, 16*wv+16); 72x V_WMMA_F32_16X16X4_F32.
//    A (16x4 f32): lane<16 -> row m=lane, K pair {kb,kb+1}; lane>=16 -> {kb+2,kb+3}.
//    B (4x16 f32): symmetric split; tw_t row-per-lane -> contiguous global_load_b64.
//  Epilogue: relu * group_mask, shfl_xor reduce over 4 rotation lanes, store.
// ---------------------------------------------------------------------------
__global__ void __launch_bounds__(256) deform_main(
    const float* __restrict__ x,     const float* __restrict__ tw_t,
    const float* __restrict__ off_y, const float* __restrict__ off_x,
    const float* __restrict__ msk,   const float* __restrict__ gmask,
    float* __restrict__ out) {
    __shared__ float sA[16 * SROW];

    int strip = blockIdx.x;                 // 0 .. B*H*(W/16)-1
    int b   = strip / (HH * (WW / 16));
    int rem = strip % (HH * (WW / 16));
    int h   = rem / (WW / 16);
    int w0  = (rem % (WW / 16)) * 16;
    int tid = threadIdx.x;

    // ---- Stage 1: bilinear sampling into LDS --------------------------------
    for (int idx = tid; idx < 16 * CK; idx += 256) {
        int m = idx / CK, ck = idx % CK;
        int c = ck / KK,  k  = ck % KK;
        int w = w0 + m;
        size_t ob = (size_t)(b * KK + k) * HH * WW + (size_t)h * WW + w;
        float oy = off_y[ob], ox = off_x[ob], mv = msk[ob];
        float py = (float)(h - PADC + k / 3) + oy;
        float px = (float)(w - PADC + k % 3) + ox;
        float fy = floorf(py), fx = floorf(px);
        float dy = py - fy,    dx = px - fx;
        int y0 = (int)fy, x0 = (int)fx;
        const float* xp = x + (size_t)(b * CIN + c) * HH * WW;
        float s = 0.f;
        {
            int yi = y0, xi = x0;
            if (yi >= 0 && yi < HH && xi >= 0 && xi < WW)
                s += xp[yi * WW + xi] * ((1.f - dy) * (1.f - dx));
        }
        {
            int yi = y0, xi = x0 + 1;
            if (yi >= 0 && yi < HH && xi >= 0 && xi < WW)
                s += xp[yi * WW + xi] * ((1.f - dy) * dx);
        }
        {
            int yi = y0 + 1, xi = x0;
            if (yi >= 0 && yi < HH && xi >= 0 && xi < WW)
                s += xp[yi * WW + xi] * (dy * (1.f - dx));
        }
        {
            int yi = y0 + 1, xi = x0 + 1;
            if (yi >= 0 && yi < HH && xi >= 0 && xi < WW)
                s += xp[yi * WW + xi] * (dy * dx);
        }
        sA[m * SROW + ck] = s * mv;
    }
    __syncthreads();

    // ---- Stage 2: f32 WMMA K-loop ------------------------------------------
    int lane  = tid & 31;
    int wv    = tid >> 5;            // N-tile id, 0..7
    int nBase = wv * 16;
    int nLoc  = lane & 15;
    int kOff  = (lane < 16) ? 0 : 2; // K-pair split across lane halves
    const float* twn  = tw_t + (size_t)(nBase + nLoc) * CK;   // B row for this lane
    const float* aRow = &sA[nLoc * SROW + kOff];              // A row (m = lane&15)

    v8f acc = {0.f, 0.f, 0.f, 0.f, 0.f, 0.f, 0.f, 0.f};
#pragma unroll 4
    for (int kb = 0; kb < CK; kb += 4) {
        v2f av = *(const v2f*)(aRow + kb);          // ds_load_b64
        v2f bv = *(const v2f*)(twn + kb + kOff);    // global_load_b64
        acc = __builtin_amdgcn_wmma_f32_16x16x4_f32(
            false, av, false, bv, (short)0, acc, false, false);
    }

    // ---- Epilogue: relu * group_mask, reduce 4 rotations, store -------------
    int o = (nBase + nLoc) >> 2;
    int r = (nBase + nLoc) & 3;
    int mHalf = (lane < 16) ? 0 : 8;
    const float* gm = gmask + ((size_t)r * HH + h) * WW + w0;
    float* op = out + ((size_t)(b * COUT + o) * HH + h) * WW + w0;
#pragma unroll
    for (int j = 0; j < 8; j++) {
        float v = acc[j];
        v = fmaxf(v, 0.f) * gm[mHalf + j];
        v += __shfl_xor(v, 1, 32);   // sum rotation pairs
        v += __shfl_xor(v, 2, 32);   // full 4-lane rotation sum
        if ((nLoc & 3) == 0) op[mHalf + j] = v;
    }
}

// ---------------------------------------------------------------------------
extern "C" void kernel_launch(void* const* d_in, const int* in_sizes, int n_in,
                              void* d_out, int out_size, void* d_ws, size_t ws_size,
                              hipStream_t stream) {
    const float* x      = (const float*)d_in[0];
    const float* weight = (const float*)d_in[1];
    const float* offw   = (const float*)d_in[2];
    const float* offb   = (const float*)d_in[3];
    const float* mskw   = (const float*)d_in[4];
    const float* mskb   = (const float*)d_in[5];
    const float* gmask  = (const float*)d_in[6];
    float* out = (float*)d_out;

    float* ws    = (float*)d_ws;
    float* tw_t  = ws;                           // 128*288        = 36864 f32
    float* off_y = tw_t + NOUT * CK;             // B*9*H*W        = 1179648 f32
    float* off_x = off_y + (size_t)BB * KK * HH * WW;
    float* msk   = off_x + (size_t)BB * KK * HH * WW;

    prep_tw<<<(NOUT * CK + 255) / 256, 256, 0, stream>>>(weight, tw_t);
    offmask_conv<<<(BB * HH * WW + 127) / 128, 128, 0, stream>>>(
        x, offw, offb, mskw, mskb, off_y, off_x, msk);
    deform_main<<<BB * HH * (WW / 16), 256, 0, stream>>>(
        x, tw_t, off_y, off_x, msk, gmask, out);
}